// TripletLoss_7559142441509
// MI455X (gfx1250) — compile-verified
//
#include <hip/hip_runtime.h>

typedef float v2f __attribute__((ext_vector_type(2)));
typedef float v8f __attribute__((ext_vector_type(8)));

#define DIM 128            // feature dim (reference: D=128)
#define SR  132            // padded LDS row stride in floats (bank-conflict avoidance)
#define WG_THREADS 64      // 2 waves (wave32)
#define ROWS_PER_WAVE 32   // 2x2 register blocking: 32x32 score tile per wave
#define ROWS_PER_WG 64     // 2 waves * 32 anchor rows
#define TILE_J 32          // column tile: two 16-wide WMMA tiles
#define KSTEPS (DIM / 4)   // 32 k-steps of the 16x16x4 fp32 WMMA
#define MARGIN 0.2f
#define NEG_INF_F (-1e30f)
#define POS_INF_F (1e30f)

__device__ __forceinline__ v8f wmma_f32(v2f a, v2f b, v8f c) {
    return __builtin_amdgcn_wmma_f32_16x16x4_f32(false, a, false, b,
                                                 (short)0, c, false, false);
}

// Load one 32x128 column tile into 16 float4 registers (64 threads).
__device__ __forceinline__ void load_tile_regs(float4* streg,
                                               const float* __restrict__ im,
                                               int j0, int t, int B) {
    const int row  = t >> 1;
    const int col0 = (t & 1) * 64;
    const float4* g = (const float4*)(im + (size_t)(j0 + row) * DIM + col0);
    #pragma unroll
    for (int q = 0; q < 16; ++q) streg[q] = g[q];
    if (j0 + TILE_J < B)  // speculative prefetch of the tile after next
        __builtin_prefetch(im + (size_t)(j0 + TILE_J + row) * DIM + col0, 0, 0);
}

__device__ __forceinline__ void store_tile_lds(float* dst, const float4* streg, int t) {
    const int row  = t >> 1;
    const int col0 = (t & 1) * 64;
    float4* s = (float4*)(dst + row * SR + col0);
    #pragma unroll
    for (int q = 0; q < 16; ++q) s[q] = streg[q];
}

__global__ void TripletZeroOut(float* out) { out[0] = 0.0f; }

__global__ __launch_bounds__(WG_THREADS)
void TripletLossFusedWmma(const float* __restrict__ im,
                          const int*   __restrict__ ids,
                          float* __restrict__ out, int B) {
    // One 33.8 KB LDS arena used twice:
    //   phase 1: 64 anchor rows (consumed into per-lane WMMA A-fragment registers)
    //   phase 2: double-buffered 32-row B tiles (2 * 32 * SR == 64 * SR floats)
    __shared__ float ldsS[2 * TILE_J * SR];

    const int t     = threadIdx.x;
    const int wave  = t >> 5;
    const int lane  = t & 31;
    const int l15   = lane & 15;
    const int half  = lane >> 4;          // 0: lanes 0-15, 1: lanes 16-31
    const int iBase = blockIdx.x * ROWS_PER_WG;
    const int waveRow0 = wave * ROWS_PER_WAVE;

    // ---- phase 1: stage A tile (64 rows x 128 floats) through LDS
    {
        const int col0 = (t & 7) * 16;
        #pragma unroll
        for (int q = 0; q < 8; ++q) {
            const int row = q * 8 + (t >> 3);
            const float4* g = (const float4*)(im + (size_t)(iBase + row) * DIM + col0);
            float4* s = (float4*)(ldsS + row * SR + col0);
            #pragma unroll
            for (int v = 0; v < 4; ++v) s[v] = g[v];
        }
    }
    __syncthreads();

    // ---- phase 2: pull this wave's A fragments into registers for the whole kernel.
    //      A 16x4 layout: lane holds rows (waveRow0 [+16]) + l15, k-cols 4*kk + 2*half + {0,1}
    v2f areg0[KSTEPS], areg1[KSTEPS];
    {
        const float* aP0 = ldsS + (waveRow0 + l15) * SR + 2 * half;
        const float* aP1 = aP0 + 16 * SR;
        #pragma unroll
        for (int kk = 0; kk < KSTEPS; ++kk) {
            areg0[kk] = *(const v2f*)(aP0 + 4 * kk);
            areg1[kk] = *(const v2f*)(aP1 + 4 * kk);
        }
    }
    __syncthreads();   // A reads done: arena now belongs to the B double-buffer

    int idrow[2][8];
    #pragma unroll
    for (int p = 0; p < 2; ++p)
        #pragma unroll
        for (int r = 0; r < 8; ++r)
            idrow[p][r] = ids[iBase + waveRow0 + 16 * p + 8 * half + r];

    float maxneg[2][8], minpos[2][8];
    #pragma unroll
    for (int p = 0; p < 2; ++p)
        #pragma unroll
        for (int r = 0; r < 8; ++r) { maxneg[p][r] = NEG_INF_F; minpos[p][r] = POS_INF_F; }

    // ---- prologue: tile 0 into buffer 0
    float4 streg[16];
    load_tile_regs(streg, im, 0, t, B);
    store_tile_lds(ldsS, streg, t);
    __syncthreads();

    const int ntiles = B / TILE_J;
    for (int tile = 0; tile < ntiles; ++tile) {
        const int cur = tile & 1;
        const int j0  = tile * TILE_J;
        const bool havenext = (tile + 1 < ntiles);

        // 1) issue global loads for the NEXT tile (latency hidden behind the WMMAs)
        if (havenext)
            load_tile_regs(streg, im, j0 + TILE_J, t, B);

        // 2) compute current tile: ds loads precede staging ds stores in program
        //    order, so the in-order LDS queue never stalls the WMMA stream.
        const float* bP0 = ldsS + cur * (TILE_J * SR) + l15 * SR + 2 * half;
        const float* bP1 = bP0 + 16 * SR;

        v8f acc00 = {0.f,0.f,0.f,0.f,0.f,0.f,0.f,0.f};
        v8f acc01 = {0.f,0.f,0.f,0.f,0.f,0.f,0.f,0.f};
        v8f acc10 = {0.f,0.f,0.f,0.f,0.f,0.f,0.f,0.f};
        v8f acc11 = {0.f,0.f,0.f,0.f,0.f,0.f,0.f,0.f};
        #pragma unroll
        for (int kk = 0; kk < KSTEPS; ++kk) {
            v2f b0 = *(const v2f*)(bP0 + 4 * kk);
            v2f b1 = *(const v2f*)(bP1 + 4 * kk);
            acc00 = wmma_f32(areg0[kk], b0, acc00);
            acc01 = wmma_f32(areg0[kk], b1, acc01);
            acc10 = wmma_f32(areg1[kk], b0, acc10);
            acc11 = wmma_f32(areg1[kk], b1, acc11);
        }

        // 3) stage the next tile into the other buffer
        if (havenext)
            store_tile_lds(ldsS + (cur ^ 1) * (TILE_J * SR), streg, t);

        // 4) fused masked row-reductions (C layout: VGPR r -> row r+8*half, lane -> col)
        const int jg0  = j0 + l15;
        const int jg1  = j0 + 16 + l15;
        const int idc0 = ids[jg0];
        const int idc1 = ids[jg1];
        #pragma unroll
        for (int p = 0; p < 2; ++p) {
            const v8f* accC0 = (p == 0) ? &acc00 : &acc10;
            const v8f* accC1 = (p == 0) ? &acc01 : &acc11;
            #pragma unroll
            for (int r = 0; r < 8; ++r) {
                const int ig = iBase + waveRow0 + 16 * p + 8 * half + r;

                float s0 = (*accC0)[r];
                if (ig == jg0) s0 = 0.0f;
                if (idrow[p][r] == idc0) { minpos[p][r] = fminf(minpos[p][r], s0); }
                else                     { maxneg[p][r] = fmaxf(maxneg[p][r], s0); }

                float s1 = (*accC1)[r];
                if (ig == jg1) s1 = 0.0f;
                if (idrow[p][r] == idc1) { minpos[p][r] = fminf(minpos[p][r], s1); }
                else                     { maxneg[p][r] = fmaxf(maxneg[p][r], s1); }
            }
        }
        __syncthreads();  // single barrier per tile (double-buffered)
    }

    // ---- reduce across the 16 lanes of each half (wave32: masks 1,2,4,8 stay in-half)
    float cost = 0.0f;
    #pragma unroll
    for (int p = 0; p < 2; ++p) {
        #pragma unroll
        for (int r = 0; r < 8; ++r) {
            float mn = maxneg[p][r], mp = minpos[p][r];
            #pragma unroll
            for (int m = 1; m <= 8; m <<= 1) {
                mn = fmaxf(mn, __shfl_xor(mn, m, 32));
                mp = fminf(mp, __shfl_xor(mp, m, 32));
            }
            cost += fmaxf(MARGIN + mn - mp, 0.0f);
        }
    }

    // lanes 0 and 16 each hold the 16-row sum of their half
    if (l15 == 0)
        atomicAdd(out, cost);
}

extern "C" void kernel_launch(void* const* d_in, const int* in_sizes, int n_in,
                              void* d_out, int out_size, void* d_ws, size_t ws_size,
                              hipStream_t stream) {
    const float* im  = (const float*)d_in[0];
    const int*   ids = (const int*)d_in[1];
    float* out = (float*)d_out;
    const int B = in_sizes[1];            // 8192 anchors; D fixed at 128

    TripletZeroOut<<<1, 1, 0, stream>>>(out);
    const int grid = B / ROWS_PER_WG;     // 128 workgroups
    TripletLossFusedWmma<<<grid, WG_THREADS, 0, stream>>>(im, ids, out, B);
}